// Splatting_81389630259628
// MI455X (gfx1250) — compile-verified
//
#include <hip/hip_runtime.h>
#include <hip/hip_bf16.h>

// Problem constants from the reference: B=8, C=4, H=512, W=960, fp32.
#define BB 8
#define CC 4
#define HH 512
#define WW 960
#define HWSZ (HH * WW)             /* 491520 */
#define NPIX (BB * HWSZ)           /* 3,932,160 pixels */
#define OUT_ELEMS (BB * CC * HWSZ) /* 15,728,640 floats in d_out */
#define EPSV 1e-07f

#define XBLK 320                   /* 10 wave32 waves; 960 = 3 * 320 exactly */

// ---------------------------------------------------------------------------
// Kernel 1: zero the accumulators (d_out channel planes + d_ws weight plane).
// float4 (b128) stores; OUT_ELEMS and NPIX are both divisible by 4.
// ---------------------------------------------------------------------------
__global__ __launch_bounds__(256) void splat_zero_kernel(float4* __restrict__ out4,
                                                         float4* __restrict__ w4) {
    const int i = blockIdx.x * blockDim.x + threadIdx.x;
    const float4 z = make_float4(0.f, 0.f, 0.f, 0.f);
    if (i < (OUT_ELEMS / 4)) out4[i] = z;
    if (i < (NPIX / 4))      w4[i]   = z;
}

// ---------------------------------------------------------------------------
// Kernel 2: forward bilinear splat with importance weighting.
// grid = (W/XBLK, H, B), block = XBLK  ->  x/y/b come free, no divisions,
// no bounds checks. Input streams are read once: NT-hinted loads keep the
// 192MB L2 dedicated to the 79MB atomic accumulation surface. Accumulation
// uses no-return global_atomic_add_f32 (unsafeAtomicAdd), resolved at the
// L2 atomic units, fire-and-forget on STOREcnt.
// ---------------------------------------------------------------------------
__global__ __launch_bounds__(XBLK) void splat_scatter_kernel(
        const float* __restrict__ frame,   // [B,4,H,W]
        const float* __restrict__ flow,    // [B,2,H,W]
        const float* __restrict__ imp,     // [B,1,H,W]
        float* __restrict__ out,           // [B,4,H,W] accumulators
        float* __restrict__ wacc) {        // [B,H,W]   weight accumulator
    const int x   = blockIdx.x * XBLK + threadIdx.x;
    const int y   = blockIdx.y;
    const int b   = blockIdx.z;
    const int rem = y * WW + x;

    const size_t flowBase = (size_t)b * 2 * HWSZ + rem;
    const float flx = __builtin_nontemporal_load(flow + flowBase);        // x disp
    const float fly = __builtin_nontemporal_load(flow + flowBase + HWSZ); // y disp
    const float e   = __expf(__builtin_nontemporal_load(imp + (size_t)b * HWSZ + rem));

    const size_t frameBase = (size_t)b * CC * HWSZ + rem;
    const float f0 = __builtin_nontemporal_load(frame + frameBase           ) * e;
    const float f1 = __builtin_nontemporal_load(frame + frameBase +     HWSZ) * e;
    const float f2 = __builtin_nontemporal_load(frame + frameBase + 2 * HWSZ) * e;
    const float f3 = __builtin_nontemporal_load(frame + frameBase + 3 * HWSZ) * e;

    const float tx  = (float)x + flx;
    const float ty  = (float)y + fly;
    const float fxf = floorf(tx);
    const float fyf = floorf(ty);
    const float dx  = tx - fxf;
    const float dy  = ty - fyf;
    const int ix = (int)fxf;
    const int iy = (int)fyf;

    const float wgt[4] = { (1.f - dx) * (1.f - dy),
                           dx         * (1.f - dy),
                           (1.f - dx) * dy,
                           dx         * dy };
    const int cxo[4] = { 0, 1, 0, 1 };
    const int cyo[4] = { 0, 0, 1, 1 };

    float* __restrict__ ob = out  + (size_t)b * CC * HWSZ;
    float* __restrict__ wb = wacc + (size_t)b * HWSZ;

#pragma unroll
    for (int k = 0; k < 4; ++k) {
        const int cx = ix + cxo[k];
        const int cy = iy + cyo[k];
        if (cx >= 0 && cx < WW && cy >= 0 && cy < HH) {
            const int   idx = cy * WW + cx;
            const float w   = wgt[k];
            unsafeAtomicAdd(ob + idx,            f0 * w);
            unsafeAtomicAdd(ob + HWSZ + idx,     f1 * w);
            unsafeAtomicAdd(ob + 2 * HWSZ + idx, f2 * w);
            unsafeAtomicAdd(ob + 3 * HWSZ + idx, f3 * w);
            unsafeAtomicAdd(wb + idx,            e  * w);
        }
    }
}

// ---------------------------------------------------------------------------
// Kernel 3: normalize: out[b,c,y,x] /= (wacc[b,y,x] + eps).
// Same divisionless mapping; one thread per pixel amortizes the weight read
// across the 4 channel planes (all lines are L2-resident from the atomics).
// ---------------------------------------------------------------------------
__global__ __launch_bounds__(XBLK) void splat_norm_kernel(float* __restrict__ out,
                                                          const float* __restrict__ wacc) {
    const int x   = blockIdx.x * XBLK + threadIdx.x;
    const int y   = blockIdx.y;
    const int b   = blockIdx.z;
    const int rem = y * WW + x;

    const float inv = 1.0f / (wacc[(size_t)b * HWSZ + rem] + EPSV);

    float* __restrict__ ob = out + (size_t)b * CC * HWSZ + rem;
    ob[0]        *= inv;
    ob[HWSZ]     *= inv;
    ob[2 * HWSZ] *= inv;
    ob[3 * HWSZ] *= inv;
}

// ---------------------------------------------------------------------------
// Launch
// ---------------------------------------------------------------------------
extern "C" void kernel_launch(void* const* d_in, const int* in_sizes, int n_in,
                              void* d_out, int out_size, void* d_ws, size_t ws_size,
                              hipStream_t stream) {
    const float* frame = (const float*)d_in[0];   // [8,4,512,960]
    const float* flow  = (const float*)d_in[1];   // [8,2,512,960]
    const float* imp   = (const float*)d_in[2];   // [8,1,512,960]
    float* out  = (float*)d_out;                  // [8,4,512,960]
    float* wacc = (float*)d_ws;                   // [8,512,960] weight plane (15.7 MB)

    (void)in_sizes; (void)n_in; (void)out_size; (void)ws_size;

    // 1) zero accumulators (d_out + weight plane); grid sized by the larger
    const int nOut4   = OUT_ELEMS / 4;            // 3,932,160 float4 stores
    const int zBlocks = (nOut4 + 255) / 256;
    splat_zero_kernel<<<zBlocks, 256, 0, stream>>>((float4*)out, (float4*)wacc);

    // 2) scatter-splat: grid (W/XBLK, H, B) = (3, 512, 8), block XBLK=320
    const dim3 grid(WW / XBLK, HH, BB);
    splat_scatter_kernel<<<grid, XBLK, 0, stream>>>(frame, flow, imp, out, wacc);

    // 3) normalize (same mapping)
    splat_norm_kernel<<<grid, XBLK, 0, stream>>>(out, wacc);
}